// GNNActorCriticModelPool_61503931678798
// MI455X (gfx1250) — compile-verified
//
#include <hip/hip_runtime.h>
#include <hip/hip_bf16.h>

// ---------------------------------------------------------------------------
// GCN (3 layers) + mean pool for fixed shapes:
//   B=16, N=10000, E=160000, Fin=10, HID=64, Fout=10
// Strategy: build CSR once (shared across batch & layers), WMMA f32 GEMMs
// with 4x register blocking in M and branchless tail padding, atomic-free
// gather aggregation; all buffers (~84 MB) are L2-resident (192 MB L2).
// ---------------------------------------------------------------------------

#define Bsz  16
#define Nn   10000
#define Ee   160000
#define FIN  10
#define HID  64
#define FOUT 10

typedef __attribute__((ext_vector_type(2))) float v2f;
typedef __attribute__((ext_vector_type(8))) float v8f;

// ---------------- graph preprocessing ----------------

__global__ void deg_count_kernel(const long long* __restrict__ ei,
                                 unsigned* __restrict__ deg) {
    int e = blockIdx.x * blockDim.x + threadIdx.x;
    if (e < Ee) {
        int d = (int)ei[Ee + e];          // dst row of edge_index
        atomicAdd(&deg[d], 1u);
    }
}

__global__ void dinv_kernel(const unsigned* __restrict__ deg,
                            float* __restrict__ dinv) {
    int n = blockIdx.x * blockDim.x + threadIdx.x;
    if (n < Nn) dinv[n] = rsqrtf((float)(deg[n] + 1u));   // +1 self loop
}

// single-block inclusive scan of (deg[n]+1) -> row_ptr (length N+1)
__global__ void scan_kernel(const unsigned* __restrict__ deg,
                            int* __restrict__ row_ptr) {
    __shared__ int buf[1024];
    int tid = threadIdx.x;
    int carry = 0;
    if (tid == 0) row_ptr[0] = 0;
    for (int base = 0; base < Nn; base += 1024) {
        int i = base + tid;
        int v = (i < Nn) ? ((int)deg[i] + 1) : 0;
        buf[tid] = v;
        __syncthreads();
        for (int off = 1; off < 1024; off <<= 1) {
            int t = (tid >= off) ? buf[tid - off] : 0;
            __syncthreads();
            buf[tid] += t;
            __syncthreads();
        }
        if (i < Nn) row_ptr[i + 1] = carry + buf[tid];
        carry += buf[1023];
        __syncthreads();
    }
}

// bin E edges + N self-loops into CSR; precompute norm = dinv[s]*dinv[d]
__global__ void fill_csr_kernel(const long long* __restrict__ ei,
                                const float* __restrict__ dinv,
                                const int* __restrict__ row_ptr,
                                int* __restrict__ cursor,
                                int* __restrict__ csr_src,
                                float* __restrict__ csr_norm) {
    int i = blockIdx.x * blockDim.x + threadIdx.x;
    if (i < Ee) {
        int s = (int)ei[i];
        int d = (int)ei[Ee + i];
        int pos = atomicAdd(&cursor[d], 1);
        int idx = row_ptr[d] + pos;
        csr_src[idx]  = s;
        csr_norm[idx] = dinv[s] * dinv[d];
    } else if (i < Ee + Nn) {
        int n = i - Ee;
        int pos = atomicAdd(&cursor[n], 1);
        int idx = row_ptr[n] + pos;
        csr_src[idx]  = n;
        csr_norm[idx] = dinv[n] * dinv[n];
    }
}

// ---------------- WMMA fp32 GEMM: H[M,NOUT] = A[M,K] @ W[K,NOUT] ------------
// One wave computes MT consecutive 16x16 output tiles in M (register
// blocking): the B fragment (weights) is loaded once per k-step and reused
// by MT independent V_WMMA_F32_16X16X4_F32 ops (independent accumulators ->
// no RAW hazards between consecutive WMMAs).
// Tail handling (K=10 / NOUT=10) is branchless: loads use in-bounds clamped
// addresses and out-of-range values are zeroed with selects (v_cndmask),
// avoiding exec save/restore around every load (EXEC must be all-1s at WMMA).
// A frag v2f: lanes 0-15 hold K=kb,kb+1; lanes 16-31 hold K=kb+2,kb+3.
// C/D v8f: VGPR r -> row r + 8*(lane>=16), col = lane&15.

template <int K, int NOUT, int MT>
__global__ __launch_bounds__(256)
void gemm_wmma_kernel(const float* __restrict__ A,
                      const float* __restrict__ W,
                      float* __restrict__ H, int Mgroups) {
    constexpr int NT = (NOUT + 15) / 16;
    int wid  = (int)((blockIdx.x * blockDim.x + threadIdx.x) >> 5);
    int lane = threadIdx.x & 31;
    int g  = wid / NT;
    int nt = wid % NT;
    if (g >= Mgroups) return;                 // wave-uniform guard

    int half = lane >> 4;                     // 0: K pair lo, 1: K pair hi
    int l    = lane & 15;
    int col  = nt * 16 + l;                   // output column for this lane
    bool cok = (col < NOUT);
    int colc = cok ? col : 0;                 // in-bounds column for loads

    const float* __restrict__ arow[MT];
    #pragma unroll
    for (int t = 0; t < MT; ++t)
        arow[t] = A + ((size_t)(g * MT + t) * 16 + l) * K;

    v8f c[MT];
    #pragma unroll
    for (int t = 0; t < MT; ++t) c[t] = (v8f){};

    #pragma unroll
    for (int kb = 0; kb < K; kb += 4) {
        int k0 = kb + 2 * half;
        bool v0 = (k0     < K);               // compile-time true except tail
        bool v1 = (k0 + 1 < K);
        int ka0 = v0 ? k0     : 0;            // in-bounds K index for loads
        int ka1 = v1 ? k0 + 1 : 0;

        v2f b;
        float bx = W[(size_t)ka0 * NOUT + colc];
        float by = W[(size_t)ka1 * NOUT + colc];
        b.x = (cok && v0) ? bx : 0.0f;
        b.y = (cok && v1) ? by : 0.0f;

        #pragma unroll
        for (int t = 0; t < MT; ++t) {
            v2f a;
            float ax = arow[t][ka0];
            float ay = arow[t][ka1];
            a.x = v0 ? ax : 0.0f;
            a.y = v1 ? ay : 0.0f;
            c[t] = __builtin_amdgcn_wmma_f32_16x16x4_f32(
                       /*neg_a=*/false, a, /*neg_b=*/false, b,
                       /*c_mod=*/(short)0, c[t],
                       /*reuse_a=*/false, /*reuse_b=*/false);
        }
    }

    if (cok) {
        #pragma unroll
        for (int t = 0; t < MT; ++t) {
            #pragma unroll
            for (int r = 0; r < 8; ++r) {
                int orow = (g * MT + t) * 16 + r + 8 * half;
                H[(size_t)orow * NOUT + col] = c[t][r];
            }
        }
    }
}

// ---------------- gather aggregation + bias + ReLU --------------------------
// block = (dst node, batch), thread = feature.  No atomics; reads of one
// source row (F*4 bytes) are fully coalesced; index/norm loads are uniform
// (scalarizable broadcast).

template <int F>
__global__ void gather_agg_kernel(const float* __restrict__ Hin,
                                  float* __restrict__ Hout,
                                  const int* __restrict__ row_ptr,
                                  const int* __restrict__ csr_src,
                                  const float* __restrict__ csr_norm,
                                  const float* __restrict__ bias) {
    int d = blockIdx.x;
    int b = blockIdx.y;
    int f = threadIdx.x;
    int beg = row_ptr[d];
    int end = row_ptr[d + 1];
    if (f < F) {
        const float* __restrict__ base = Hin + (size_t)b * Nn * F;
        float acc = 0.0f;
        for (int i = beg; i < end; ++i) {
            int   s = csr_src[i];
            float w = csr_norm[i];
            acc += w * base[(size_t)s * F + f];
        }
        float v = acc + bias[f];
        Hout[((size_t)b * Nn + d) * F + f] = v > 0.0f ? v : 0.0f;
    }
}

// ---------------- global mean pool over nodes -------------------------------

__global__ void pool_kernel(const float* __restrict__ A,
                            float* __restrict__ out) {
    __shared__ float sbuf[256];
    int b = blockIdx.x;
    int tid = threadIdx.x;
    float acc[FOUT];
    #pragma unroll
    for (int f = 0; f < FOUT; ++f) acc[f] = 0.0f;
    const float* __restrict__ base = A + (size_t)b * Nn * FOUT;
    for (int n = tid; n < Nn; n += 256) {
        const float* r = base + (size_t)n * FOUT;
        #pragma unroll
        for (int f = 0; f < FOUT; ++f) acc[f] += r[f];
    }
    for (int f = 0; f < FOUT; ++f) {
        sbuf[tid] = acc[f];
        __syncthreads();
        for (int s = 128; s > 0; s >>= 1) {
            if (tid < s) sbuf[tid] += sbuf[tid + s];
            __syncthreads();
        }
        if (tid == 0) out[b * FOUT + f] = sbuf[0] * (1.0f / (float)Nn);
        __syncthreads();
    }
}

// ---------------- host launcher ---------------------------------------------

extern "C" void kernel_launch(void* const* d_in, const int* in_sizes, int n_in,
                              void* d_out, int out_size, void* d_ws, size_t ws_size,
                              hipStream_t stream) {
    const float*     x  = (const float*)d_in[0];
    const long long* ei = (const long long*)d_in[1];   // int64 [2,E]
    const float* W1 = (const float*)d_in[2];
    const float* b1 = (const float*)d_in[3];
    const float* W2 = (const float*)d_in[4];
    const float* b2 = (const float*)d_in[5];
    const float* W3 = (const float*)d_in[6];
    const float* b3 = (const float*)d_in[7];
    float* out = (float*)d_out;

    // workspace layout (bytes)
    char* ws = (char*)d_ws;
    const size_t SZ_BUF = (size_t)Bsz * Nn * HID * sizeof(float); // 40,960,000
    float*    bufA     = (float*)(ws);
    float*    bufB     = (float*)(ws + SZ_BUF);
    unsigned* deg      = (unsigned*)(ws + 2 * SZ_BUF);
    float*    dinv     = (float*)(ws + 2 * SZ_BUF +   40192);
    int*      row_ptr  = (int*)(ws + 2 * SZ_BUF +   80384);
    int*      cursor   = (int*)(ws + 2 * SZ_BUF +  120832);
    int*      csr_src  = (int*)(ws + 2 * SZ_BUF +  161024);
    float*    csr_norm = (float*)(ws + 2 * SZ_BUF +  841024);
    (void)in_sizes; (void)n_in; (void)out_size; (void)ws_size;

    // --- graph preprocessing (cheap; redone every call for determinism) ---
    hipMemsetAsync(deg,    0, Nn * sizeof(unsigned), stream);
    hipMemsetAsync(cursor, 0, Nn * sizeof(int),      stream);
    deg_count_kernel<<<(Ee + 255) / 256, 256, 0, stream>>>(ei, deg);
    dinv_kernel<<<(Nn + 255) / 256, 256, 0, stream>>>(deg, dinv);
    scan_kernel<<<1, 1024, 0, stream>>>(deg, row_ptr);
    fill_csr_kernel<<<(Ee + Nn + 255) / 256, 256, 0, stream>>>(
        ei, dinv, row_ptr, cursor, csr_src, csr_norm);

    const int MT      = 4;
    const int Mtiles  = (Bsz * Nn) / 16;           // 10000
    const int Mgroups = Mtiles / MT;               // 2500

    // layer 1: [B*N,10] @ [10,64]
    {
        int waves  = Mgroups * ((HID + 15) / 16);  // 10000
        int blocks = (waves + 7) / 8;              // 256 thr = 8 waves
        gemm_wmma_kernel<FIN, HID, 4><<<blocks, 256, 0, stream>>>(x, W1, bufA, Mgroups);
        gather_agg_kernel<HID><<<dim3(Nn, Bsz), 64, 0, stream>>>(
            bufA, bufB, row_ptr, csr_src, csr_norm, b1);
    }
    // layer 2: [B*N,64] @ [64,64]
    {
        int waves  = Mgroups * ((HID + 15) / 16);
        int blocks = (waves + 7) / 8;
        gemm_wmma_kernel<HID, HID, 4><<<blocks, 256, 0, stream>>>(bufB, W2, bufA, Mgroups);
        gather_agg_kernel<HID><<<dim3(Nn, Bsz), 64, 0, stream>>>(
            bufA, bufB, row_ptr, csr_src, csr_norm, b2);
    }
    // layer 3: [B*N,64] @ [64,10]
    {
        int waves  = Mgroups * ((FOUT + 15) / 16); // 2500
        int blocks = (waves + 7) / 8;              // 313
        gemm_wmma_kernel<HID, FOUT, 4><<<blocks, 256, 0, stream>>>(bufB, W3, bufA, Mgroups);
        gather_agg_kernel<FOUT><<<dim3(Nn, Bsz), 32, 0, stream>>>(
            bufA, bufB, row_ptr, csr_src, csr_norm, b3);
    }
    // mean pool
    pool_kernel<<<Bsz, 256, 0, stream>>>(bufB, out);
}